// QuadraticCrossEntropyLoss_59811714564769
// MI455X (gfx1250) — compile-verified
//
#include <hip/hip_runtime.h>

typedef __attribute__((ext_vector_type(2))) float v2f;
typedef __attribute__((ext_vector_type(8))) float v8f;

// ---------------------------------------------------------------------------
// Kernel 1: per-batch softmax + cross-entropy.
//   grid = B blocks, 256 threads. Writes p[b][N] to workspace, ce[b].
// ---------------------------------------------------------------------------
__global__ void softmax_ce_kernel(const float* __restrict__ m,
                                  const int*   __restrict__ targets,
                                  float* __restrict__ p,
                                  float* __restrict__ ce,
                                  int N) {
    const int b   = blockIdx.x;
    const int tid = threadIdx.x;
    const int T   = blockDim.x;          // 256
    const float* mb = m + (size_t)b * N;
    float*       pb = p + (size_t)b * N;

    __shared__ float red[256];

    // pass 1: row max
    float mx = -3.402823466e+38f;
    for (int n = tid; n < N; n += T) mx = fmaxf(mx, mb[n]);
    red[tid] = mx;
    __syncthreads();
    for (int s = T / 2; s > 0; s >>= 1) {
        if (tid < s) red[tid] = fmaxf(red[tid], red[tid + s]);
        __syncthreads();
    }
    mx = red[0];
    __syncthreads();

    // pass 2: exp + sum (keep exps in registers, N/T = 4)
    float e[8];
    float sum = 0.f;
    int cnt = 0;
    for (int n = tid; n < N; n += T) {
        float v = expf(mb[n] - mx);
        e[cnt++] = v;
        sum += v;
    }
    red[tid] = sum;
    __syncthreads();
    for (int s = T / 2; s > 0; s >>= 1) {
        if (tid < s) red[tid] += red[tid + s];
        __syncthreads();
    }
    sum = red[0];

    const float inv = 1.0f / sum;
    cnt = 0;
    for (int n = tid; n < N; n += T) pb[n] = e[cnt++] * inv;

    if (tid == 0) {
        const int t = targets[b];
        ce[b] = mx + logf(sum) - mb[t];
    }
}

// ---------------------------------------------------------------------------
// Kernel 2: streaming WMMA strips.
//   grid = (N/16/8, B), 256 threads (8 waves). Wave w handles 16 rows of K[b]:
//   d[i] = sum_j K[i,j] p[j] via v_wmma_f32_16x16x4_f32 accumulation.
//   B128 streaming: per 8 K-columns, lanes 0-15 carry cols j..j+3 and lanes
//   16-31 carry cols j+4..j+7 (one global_load_b128); the two A operands are
//   VGPR pairs (0,1) and (2,3). One ds_load_b128 of p with the same per-lane
//   column base supplies both matching B operands (p broadcast across the 16
//   output columns). K-dim ordering inside the WMMA is virtual; A/B agree.
//   Then partial[b][strip] = 0.5*(diag_part - sum_i p[i] d[i]).
// ---------------------------------------------------------------------------
__global__ void quad_diag_kernel(const float* __restrict__ k,
                                 const float* __restrict__ p,
                                 float* __restrict__ part,
                                 int N) {
    const int b     = blockIdx.y;
    const int wave  = threadIdx.x >> 5;
    const int lane  = threadIdx.x & 31;
    const int strip = blockIdx.x * (blockDim.x >> 5) + wave;  // 0 .. N/16-1
    const int n0    = strip * 16;

    extern __shared__ float sp[];  // N floats: p[b][:]
    const float* pb = p + (size_t)b * N;
    for (int i = threadIdx.x * 4; i < N; i += blockDim.x * 4) {
        *(float4*)&sp[i] = *(const float4*)&pb[i];
    }
    __syncthreads();

    const float* kb   = k + (size_t)b * N * N;
    const int    row  = n0 + (lane & 15);
    const int    coff = (lane >> 4) << 2;  // 0 for lanes 0-15, 4 for 16-31
    const float* arow = kb + (size_t)row * N + coff;

    v8f c = {0.f, 0.f, 0.f, 0.f, 0.f, 0.f, 0.f, 0.f};

#pragma unroll 4
    for (int j = 0; j < N; j += 8) {
        if ((j & 31) == 0) {
            // keep the HBM stream ahead: speculative prefetch ~1KB down the row
            __builtin_prefetch(arow + j + 256, 0, 0);
        }
        // A: K[row][j+coff .. j+coff+3]  (16B contiguous per lane)
        float4 kv = *(const float4*)(arow + j);
        // B: p[j+coff .. j+coff+3] from LDS (broadcast across output columns)
        float4 pv = *(const float4*)&sp[j + coff];

        v2f a0;  a0.x = kv.x;  a0.y = kv.y;
        v2f b0;  b0.x = pv.x;  b0.y = pv.y;
        c = __builtin_amdgcn_wmma_f32_16x16x4_f32(
                false, a0, false, b0, (short)0, c, false, false);

        v2f a1;  a1.x = kv.z;  a1.y = kv.w;
        v2f b1;  b1.x = pv.z;  b1.y = pv.w;
        c = __builtin_amdgcn_wmma_f32_16x16x4_f32(
                false, a1, false, b1, (short)0, c, false, false);
    }

    // quad partial: each half-wave holds d[rowbase..rowbase+7] replicated
    const int rowbase = (lane & 16) ? 8 : 0;
    float s = 0.f;
#pragma unroll
    for (int v = 0; v < 8; ++v) s += c[v] * sp[n0 + rowbase + v];
    s += __shfl_xor(s, 16, 32);   // s = sum_i p[n0+i] * d[i] on every lane

    // diag partial for rows n0..n0+15
    float d = 0.f;
    if (lane < 16) {
        const int n = n0 + lane;
        d = kb[(size_t)n * N + n] * sp[n];
    }
#pragma unroll
    for (int off = 16; off > 0; off >>= 1) d += __shfl_xor(d, off, 32);

    if (lane == 0) {
        part[(size_t)b * (N / 16) + strip] = 0.5f * (d - s);
    }
}

// ---------------------------------------------------------------------------
// Kernel 3: combine. out[b] = ce[b] + sum_s partial[b][s]
// ---------------------------------------------------------------------------
__global__ void combine_kernel(const float* __restrict__ ce,
                               const float* __restrict__ part,
                               float* __restrict__ out,
                               int B, int S) {
    const int b = blockIdx.x * blockDim.x + threadIdx.x;
    if (b >= B) return;
    float acc = ce[b];
    const float* pp = part + (size_t)b * S;
    for (int i = 0; i < S; ++i) acc += pp[i];
    out[b] = acc;
}

extern "C" void kernel_launch(void* const* d_in, const int* in_sizes, int n_in,
                              void* d_out, int out_size, void* d_ws, size_t ws_size,
                              hipStream_t stream) {
    const float* m       = (const float*)d_in[0];
    const float* k       = (const float*)d_in[1];
    const int*   targets = (const int*)d_in[2];
    float*       out     = (float*)d_out;

    const int B = in_sizes[2];              // 128
    const int N = in_sizes[0] / B;          // 1024
    const int S = N / 16;                   // strips per batch (64)

    float* ws_p    = (float*)d_ws;                  // B*N floats
    float* ws_ce   = ws_p  + (size_t)B * N;         // B floats
    float* ws_part = ws_ce + B;                     // B*S floats

    softmax_ce_kernel<<<B, 256, 0, stream>>>(m, targets, ws_p, ws_ce, N);

    dim3 grid(S / 8, B);                    // 8 waves per block -> 8 strips
    quad_diag_kernel<<<grid, 256, N * sizeof(float), stream>>>(k, ws_p, ws_part, N);

    combine_kernel<<<(B + 127) / 128, 128, 0, stream>>>(ws_ce, ws_part, out, B, S);
}